// KLDivFullGaussian_78185584656642
// MI455X (gfx1250) — compile-verified
//
#include <hip/hip_runtime.h>
#include <math.h>

// ---------------------------------------------------------------------------
// KL( N(p_mu,p_cov) || N(q_mu,q_cov) ), full 64x64 covariances, B=16384.
//   q = Lq Lq^T, p = Lp Lp^T (Cholesky)
//   tr(q^-1 p)   = || Lq^-1 Lp ||_F^2
//   maha         = || Lq^-1 (q_mu - p_mu) ||^2
//   logdet       = 2 * sum(log diag(L))
// One wave32 per batch item; matrices live in LDS; 16x16-blocked factorization
// and triangular solve with v_wmma_f32_16x16x4_f32 doing all block-GEMM work.
// Global->LDS input streaming uses GLOBAL_LOAD_ASYNC_TO_LDS_B128 (ASYNCcnt)
// when the toolchain exposes it, else b128 VGPR staging with NT loads.
// ---------------------------------------------------------------------------

typedef __attribute__((ext_vector_type(2))) float v2f;
typedef __attribute__((ext_vector_type(4))) float v4f;
typedef __attribute__((ext_vector_type(8))) float v8f;
typedef __attribute__((ext_vector_type(4))) int   v4i;

typedef __attribute__((address_space(1))) v4i glb_v4i;
typedef __attribute__((address_space(3))) v4i lds_v4i;

#define LS 68  // LDS row stride (floats); 68*4 = 272 bytes = 16-byte aligned rows

#if defined(__has_builtin)
#if __has_builtin(__builtin_amdgcn_global_load_async_to_lds_b128)
#define USE_ASYNC_LDS 1
#endif
#if __has_builtin(__builtin_amdgcn_s_wait_asynccnt)
#define HAVE_WAIT_ASYNC_BUILTIN 1
#endif
#endif

__device__ __forceinline__ void wait_async_zero() {
#if defined(HAVE_WAIT_ASYNC_BUILTIN)
  __builtin_amdgcn_s_wait_asynccnt(0);
#else
  asm volatile("s_wait_asynccnt 0" ::: "memory");
#endif
}

__device__ __forceinline__ float wave_sum32(float v) {
#pragma unroll
  for (int o = 16; o > 0; o >>= 1) v += __shfl_down(v, o, 32);
  return v;  // total lands in lane 0
}

// ---- WMMA fragment gathers (CDNA5 ISA 7.12.2 layouts, wave32) --------------
// C/D 16x16 f32: VGPR t -> row t (lanes 0-15) / row t+8 (lanes 16-31), col = lane&15
__device__ __forceinline__ v8f load_cfrag(const float* M, int r0, int c0, int lane) {
  const int col = c0 + (lane & 15);
  const int rb  = r0 + ((lane >> 4) << 3);
  v8f c;
#pragma unroll
  for (int t = 0; t < 8; ++t) c[t] = M[(rb + t) * LS + col];
  return c;
}
__device__ __forceinline__ void store_cfrag(float* M, int r0, int c0, int lane, v8f c) {
  const int col = c0 + (lane & 15);
  const int rb  = r0 + ((lane >> 4) << 3);
#pragma unroll
  for (int t = 0; t < 8; ++t) M[(rb + t) * LS + col] = c[t];
}
// A 16x4 f32: row = lane&15; VGPR0 holds K = 0(lanes 0-15)/2(lanes 16-31), VGPR1 K = 1/3
__device__ __forceinline__ v2f load_afrag(const float* M, int r0, int c0, int lane) {
  const int r = r0 + (lane & 15);
  const int k = c0 + ((lane >> 4) << 1);
  v2f a;
  a.x = M[r * LS + k];
  a.y = M[r * LS + k + 1];
  return a;
}
__device__ __forceinline__ v2f load_afrag_neg(const float* M, int r0, int c0, int lane) {
  v2f a = load_afrag(M, r0, c0, lane);
  a.x = -a.x;
  a.y = -a.y;
  return a;
}
// B 4x16 f32 (mirror of A): col = lane&15; VGPR0 K = 0/2, VGPR1 K = 1/3
__device__ __forceinline__ v2f load_bfrag(const float* M, int r0, int c0, int lane) {
  const int n = c0 + (lane & 15);
  const int k = r0 + ((lane >> 4) << 1);
  v2f b;
  b.x = M[k * LS + n];
  b.y = M[(k + 1) * LS + n];
  return b;
}

__device__ __forceinline__ v8f wmma_acc(v2f a, v2f b, v8f c) {
  // (neg_a, A, neg_b, B, c_mod, C, reuse_a, reuse_b)
  return __builtin_amdgcn_wmma_f32_16x16x4_f32(false, a, false, b, (short)0, c,
                                               false, false);
}

// C(16x16 at ci,cj) -= Lsrc(ci,K0) * Lsrc(cj,K0)^T   (SYRK-style trailing update)
__device__ __forceinline__ void syrk_update(float* C, int ciR, int cjR,
                                            const float* Lsrc, int K0, int lane) {
  v8f c = load_cfrag(C, ciR, cjR, lane);
#pragma unroll
  for (int kk = 0; kk < 4; ++kk) {
    v2f a = load_afrag_neg(Lsrc, ciR, K0 + 4 * kk, lane);
    // B = (L block at cjR,K0)^T -> same gather pattern as an A fragment
    v2f b = load_afrag(Lsrc, cjR, K0 + 4 * kk, lane);
    c = wmma_acc(a, b, c);
  }
  store_cfrag(C, ciR, cjR, lane, c);
}

// Unblocked 16x16 Cholesky on diagonal block at (K0,K0); lanes 0-15 own rows.
__device__ __forceinline__ void potrf16(float* A, int K0, int lane) {
  const int r = lane & 15;
  const bool act = lane < 16;
  for (int j = 0; j < 16; ++j) {
    float dj  = A[(K0 + j) * LS + (K0 + j)];
    float ljj = sqrtf(dj);
    float inv = 1.0f / ljj;
    if (act && r == j) A[(K0 + j) * LS + (K0 + j)] = ljj;
    if (act && r > j) A[(K0 + r) * LS + (K0 + j)] *= inv;
    __syncthreads();
    if (act && r > j) {
      float lrj = A[(K0 + r) * LS + (K0 + j)];
      for (int c = j + 1; c <= r; ++c)
        A[(K0 + r) * LS + (K0 + c)] -= lrj * A[(K0 + c) * LS + (K0 + j)];
    }
    __syncthreads();
  }
}

// Row-parallel TRSM: rows R0..R0+15 of panel solve X * Lkk^T = A_panel.
__device__ __forceinline__ void trsm_rows16(float* A, int R0, int K0, int r) {
  for (int j = 0; j < 16; ++j) {
    float v = A[(R0 + r) * LS + (K0 + j)];
    for (int m = 0; m < j; ++m)
      v -= A[(R0 + r) * LS + (K0 + m)] * A[(K0 + j) * LS + (K0 + m)];
    A[(R0 + r) * LS + (K0 + j)] = v / A[(K0 + j) * LS + (K0 + j)];
  }
}

// Blocked (16x16) right-looking Cholesky of a 64x64 SPD matrix in LDS.
__device__ void cholesky64(float* A, int lane) {
  for (int kb = 0; kb < 4; ++kb) {
    const int K0 = kb * 16;
    potrf16(A, K0, lane);
    __syncthreads();
    // panels: two at a time (lanes 0-15 -> ib, lanes 16-31 -> ib+1)
    for (int ib = kb + 1; ib < 4; ib += 2) {
      const int myib = ib + (lane >> 4);
      if (myib < 4) trsm_rows16(A, myib * 16, K0, lane & 15);
    }
    __syncthreads();
    // WMMA trailing updates
    for (int jb = kb + 1; jb < 4; ++jb)
      for (int ib = jb; ib < 4; ++ib)
        syrk_update(A, ib * 16, jb * 16, A, K0, lane);
    __syncthreads();
  }
}

__global__ void __launch_bounds__(32)
kl_full_gaussian_kernel(const float* __restrict__ p_mu, const float* __restrict__ p_cov,
                        const float* __restrict__ q_mu, const float* __restrict__ q_cov,
                        float* __restrict__ out) {
  __shared__ float Aq[64 * LS];  // q_cov -> Lq
  __shared__ float Ap[64 * LS];  // p_cov -> Lp -> W = Lq^-1 Lp
  __shared__ float xv[64];       // diff -> forward-substitution state

  const int lane = threadIdx.x;
  const long b = blockIdx.x;

  // ---- stream in both covariances (read-once stream, 537MB total) -----------
  const v4f* gq = reinterpret_cast<const v4f*>(q_cov + (size_t)b * 4096);
  const v4f* gp = reinterpret_cast<const v4f*>(p_cov + (size_t)b * 4096);
#if defined(USE_ASYNC_LDS)
  // Fire-and-forget DMA into LDS via ASYNCcnt path (no VGPR staging).
  for (int i = lane; i < 1024; i += 32) {
    const int r = i >> 4;         // element index e = 4i; row = e >> 6
    const int c = (i << 2) & 63;  // col = e & 63 (multiple of 4 -> 16B aligned)
    __builtin_amdgcn_global_load_async_to_lds_b128(
        (glb_v4i*)&gq[i], (lds_v4i*)&Aq[r * LS + c], 0, 0);
    __builtin_amdgcn_global_load_async_to_lds_b128(
        (glb_v4i*)&gp[i], (lds_v4i*)&Ap[r * LS + c], 0, 0);
  }
  wait_async_zero();
#else
  for (int i = lane; i < 1024; i += 32) {
    v4f vq = __builtin_nontemporal_load(&gq[i]);
    v4f vp = __builtin_nontemporal_load(&gp[i]);
    const int r = i >> 4;
    const int c = (i << 2) & 63;
    *reinterpret_cast<v4f*>(&Aq[r * LS + c]) = vq;
    *reinterpret_cast<v4f*>(&Ap[r * LS + c]) = vp;
  }
#endif
  xv[lane]      = q_mu[b * 64 + lane]      - p_mu[b * 64 + lane];
  xv[lane + 32] = q_mu[b * 64 + lane + 32] - p_mu[b * 64 + lane + 32];
  __syncthreads();

  // ---- Cholesky factorizations (WMMA trailing updates) ----------------------
  cholesky64(Aq, lane);
  cholesky64(Ap, lane);
  __syncthreads();

  // ---- log-determinants (capture diagonals before W overwrites Ap) ----------
  float ld = logf(Aq[lane * LS + lane]) + logf(Aq[(lane + 32) * LS + (lane + 32)]);
  float lp = logf(Ap[lane * LS + lane]) + logf(Ap[(lane + 32) * LS + (lane + 32)]);
  float logdet_q = 2.0f * wave_sum32(ld);  // valid in lane 0
  float logdet_p = 2.0f * wave_sum32(lp);  // valid in lane 0

  // ---- Mahalanobis: column-sweep forward substitution with Lq ---------------
  float maha = 0.0f;  // identical in every lane
  for (int k = 0; k < 64; ++k) {
    __syncthreads();
    const float xk = xv[k] / Aq[k * LS + k];
    maha += xk * xk;
    for (int i = k + 1 + lane; i < 64; i += 32)
      xv[i] -= Aq[i * LS + k] * xk;
  }
  __syncthreads();

  // ---- zero upper triangles of Ap's diagonal blocks (W is lower-tri) --------
  for (int I = 0; I < 4; ++I) {
    const int base = I * 16;
    for (int t = lane; t < 256; t += 32) {
      const int r = t >> 4, c = t & 15;
      if (c > r) Ap[(base + r) * LS + (base + c)] = 0.0f;
    }
  }
  __syncthreads();

  // ---- W = Lq^-1 Lp, blocked forward substitution; tr = ||W||_F^2 -----------
  float frob = 0.0f;
  for (int I = 0; I < 4; ++I) {
    const int IR = I * 16;
    for (int J = 0; J <= I; ++J) {
      const int JR = J * 16;
      if (J < I) {
        v8f c = load_cfrag(Ap, IR, JR, lane);
        for (int K2 = J; K2 < I; ++K2) {
          const int KC = K2 * 16;
#pragma unroll
          for (int kk = 0; kk < 4; ++kk) {
            v2f a  = load_afrag_neg(Aq, IR, KC + 4 * kk, lane);  // -Lq[I][K2]
            v2f bf = load_bfrag(Ap, KC + 4 * kk, JR, lane);      //  W[K2][J]
            c = wmma_acc(a, bf, c);
          }
        }
        store_cfrag(Ap, IR, JR, lane, c);
      }
      __syncthreads();
      // triangular solve Lq[I][I] * X = C; lanes 0-15 own columns
      if (lane < 16) {
        const int n = JR + lane;
        for (int r = 0; r < 16; ++r) {
          float v = Ap[(IR + r) * LS + n];
          for (int m = 0; m < r; ++m)
            v -= Aq[(IR + r) * LS + (IR + m)] * Ap[(IR + m) * LS + n];
          Ap[(IR + r) * LS + n] = v / Aq[(IR + r) * LS + (IR + r)];
        }
      }
      __syncthreads();
      // accumulate Frobenius norm of this W block
      {
        const int col = JR + (lane & 15);
        const int rb  = IR + ((lane >> 4) << 3);
#pragma unroll
        for (int t = 0; t < 8; ++t) {
          const float e = Ap[(rb + t) * LS + col];
          frob += e * e;
        }
      }
    }
  }
  const float tr = wave_sum32(frob);  // valid in lane 0

  if (lane == 0)
    out[b] = 0.5f * (tr + maha + logdet_q - logdet_p - 64.0f);
}

extern "C" void kernel_launch(void* const* d_in, const int* in_sizes, int n_in,
                              void* d_out, int out_size, void* d_ws, size_t ws_size,
                              hipStream_t stream) {
  const float* p_mu  = (const float*)d_in[0];
  const float* p_cov = (const float*)d_in[1];
  const float* q_mu  = (const float*)d_in[2];
  const float* q_cov = (const float*)d_in[3];
  float* out = (float*)d_out;
  const int B = in_sizes[0] / 64;  // p_mu is [B,64]
  kl_full_gaussian_kernel<<<B, 32, 0, stream>>>(p_mu, p_cov, q_mu, q_cov, out);
}